// AttentionModule_59854664237662
// MI455X (gfx1250) — compile-verified
//
#include <hip/hip_runtime.h>
#include <hip/hip_bf16.h>

// Problem constants (from reference)
#define NTOT      16384
#define SPLIT     100
#define IDIM      128
#define NHEAD     4
#define PERD      64
#define ODIM      (NHEAD * PERD)      // 256
#define ROWLEN    (SPLIT * IDIM)      // 12800 floats per sample
#define NEG_SLOPE 0.2f

#define SAMP_PER_WG 16
#define THREADS     256

// LDS layout (floats), X rows padded 128->132 to avoid bank conflicts,
// Y padded per-sample 512->516 so WMMA A-frag reads hit distinct banks.
#define XSTRIDE   132
#define XBUF_F    (SPLIT * XSTRIDE)         // 13200
#define YSTRIDE   516
#define SM_X0     0
#define SM_X1     (SM_X0 + XBUF_F)
#define SM_V      (SM_X1 + XBUF_F)          // 26400
#define SM_SCORE  (SM_V + NHEAD * IDIM)     // +512
#define SM_ATT    (SM_SCORE + NHEAD * 104)  // +416
#define SM_Y      (SM_ATT + NHEAD * 104)    // +416
#define SM_TOTALF (SM_Y + SAMP_PER_WG * YSTRIDE)  // +8256 = 36000 floats
#define SM_BYTES  (SM_TOTALF * 4)           // 144000 B < 160KB -> 2 WG/WGP

typedef __attribute__((ext_vector_type(2))) float v2f;
typedef __attribute__((ext_vector_type(8))) float v8f;

// gfx1250 async global->LDS copy path (ASYNCcnt-tracked), guarded so the
// register-pipeline fallback keeps the kernel valid on toolchains without it.
#if defined(__has_builtin)
#if __has_builtin(__builtin_amdgcn_global_load_async_to_lds_b64) && \
    __has_builtin(__builtin_amdgcn_s_wait_asynccnt)
#define USE_ASYNC_LDS 1
#endif
#endif

#ifdef USE_ASYNC_LDS
// Exact parameter types per hipcc diagnostic:
//   param0: '__attribute__((__vector_size__(2*sizeof(int)))) int __device__ *'
//           (int2 vector in addrspace(1))
//   param1: int2 vector in addrspace(3)
typedef int v2i __attribute__((vector_size(2 * sizeof(int))));
typedef __attribute__((address_space(1))) v2i g_v2i;
typedef __attribute__((address_space(3))) v2i l_v2i;

// Issue one 51.2KB sample tile as 25 per-lane b64 async copies (all 256
// threads participate -> 6400 float2 = 100x128 floats), padded-row layout.
__device__ __forceinline__ void async_copy_sample(const float* __restrict__ xg,
                                                  float* Xb, int t) {
#pragma unroll
    for (int j = 0; j < 25; ++j) {
        int off = 2 * (t + 256 * j);
        int s = off >> 7;
        int c = off & 127;
        __builtin_amdgcn_global_load_async_to_lds_b64(
            (g_v2i*)(xg + off),
            (l_v2i*)(Xb + s * XSTRIDE + c),
            /*offset=*/0, /*cpol=*/0);
    }
}
#endif

// ---------------------------------------------------------------------------
// Kernel A: v[h][i] = sum_o W[h][i][o] * q[h][o]   (4*128 outputs)
// ---------------------------------------------------------------------------
__global__ void compute_v_kernel(const float* __restrict__ W,
                                 const float* __restrict__ q,
                                 float* __restrict__ v) {
    int t = threadIdx.x;                 // 0..511
    int h = t >> 7;
    int i = t & 127;
    const float* Wr = W + ((size_t)h * IDIM + i) * PERD;
    const float* qr = q + h * PERD;
    float acc = 0.f;
#pragma unroll
    for (int o = 0; o < PERD; ++o) acc = fmaf(Wr[o], qr[o], acc);
    v[t] = acc;
}

// ---------------------------------------------------------------------------
// Main fused kernel: scores -> softmax -> y accumulation -> WMMA y@W
// One workgroup = 16 samples, 256 threads = 8 wave32s.
// ---------------------------------------------------------------------------
__global__ void __launch_bounds__(THREADS)
attn_pool_kernel(const float* __restrict__ x,
                 const float* __restrict__ W,
                 const float* __restrict__ vG,
                 float* __restrict__ out) {
    extern __shared__ float sm[];
    float* Xs0    = sm + SM_X0;
    float* Xs1    = sm + SM_X1;
    float* sv     = sm + SM_V;
    float* sScore = sm + SM_SCORE;
    float* sAtt   = sm + SM_ATT;
    float* sY     = sm + SM_Y;

    const int t  = threadIdx.x;
    const int n0 = blockIdx.x * SAMP_PER_WG;

    // stage v (4x128) into LDS
    sv[t]       = vG[t];
    sv[t + 256] = vG[t + 256];

#ifdef USE_ASYNC_LDS
    // prologue: start sample 0 copy (ASYNCcnt += 25 per wave)
    async_copy_sample(x + (size_t)n0 * ROWLEN, Xs0, t);
#else
    // register pipeline buffer: 25 float2 per thread == one 51.2KB sample tile
    float2 reg[25];
    {
        const float2* p = (const float2*)(x + (size_t)n0 * ROWLEN);
#pragma unroll
        for (int j = 0; j < 25; ++j) reg[j] = p[t + 256 * j];
    }
#endif

    for (int k = 0; k < SAMP_PER_WG; ++k) {
        float* Xb = (k & 1) ? Xs1 : Xs0;

        __syncthreads();   // prev iter's reads of buf[(k+1)&1] done; v visible

#ifdef USE_ASYNC_LDS
        // kick off sample k+1 into the other buffer, then release only the
        // older group (async loads complete in order per wave).
        if (k + 1 < SAMP_PER_WG) {
            float* Xn = ((k + 1) & 1) ? Xs1 : Xs0;
            async_copy_sample(x + (size_t)(n0 + k + 1) * ROWLEN, Xn, t);
            __builtin_amdgcn_s_wait_asynccnt(25);   // sample k's 25 are done
        } else {
            __builtin_amdgcn_s_wait_asynccnt(0);
        }
#else
        // registers -> LDS (padded rows)
#pragma unroll
        for (int j = 0; j < 25; ++j) {
            int off = 2 * (t + 256 * j);
            int s = off >> 7;
            int c = off & 127;
            *(float2*)&Xb[s * XSTRIDE + c] = reg[j];
        }
        // prefetch next sample into registers; the s_wait lands at the
        // ds_store of the next iteration, overlapping all compute below.
        if (k + 1 < SAMP_PER_WG) {
            const float2* p = (const float2*)(x + (size_t)(n0 + k + 1) * ROWLEN);
#pragma unroll
            for (int j = 0; j < 25; ++j) reg[j] = p[t + 256 * j];
        }
#endif

        __syncthreads();   // all waves' portions of X tile visible

        // ---- scores: score[h][s] = leaky_relu( x[s,:] . v[h,:] ) ----
        if (t < SPLIT) {
            const float* xr = &Xb[t * XSTRIDE];
            float a0 = 0.f, a1 = 0.f, a2 = 0.f, a3 = 0.f;
#pragma unroll 8
            for (int i = 0; i < IDIM; ++i) {
                float xv = xr[i];
                a0 = fmaf(xv, sv[i],           a0);
                a1 = fmaf(xv, sv[IDIM + i],    a1);
                a2 = fmaf(xv, sv[2*IDIM + i],  a2);
                a3 = fmaf(xv, sv[3*IDIM + i],  a3);
            }
            sScore[0 * 104 + t] = (a0 > 0.f) ? a0 : NEG_SLOPE * a0;
            sScore[1 * 104 + t] = (a1 > 0.f) ? a1 : NEG_SLOPE * a1;
            sScore[2 * 104 + t] = (a2 > 0.f) ? a2 : NEG_SLOPE * a2;
            sScore[3 * 104 + t] = (a3 > 0.f) ? a3 : NEG_SLOPE * a3;
        }
        __syncthreads();

        // ---- softmax over the 100 splits, one wave per head ----
        {
            int wid  = t >> 5;
            int lane = t & 31;
            if (wid < NHEAD) {
                const float* sb = &sScore[wid * 104];
                float s0 = sb[lane];
                float s1 = sb[lane + 32];
                float s2 = sb[lane + 64];
                float s3 = (lane < 4) ? sb[lane + 96] : -3.4e38f;
                float mx = fmaxf(fmaxf(s0, s1), fmaxf(s2, s3));
#pragma unroll
                for (int off = 16; off >= 1; off >>= 1)
                    mx = fmaxf(mx, __shfl_xor(mx, off, 32));
                float e0 = __expf(s0 - mx);
                float e1 = __expf(s1 - mx);
                float e2 = __expf(s2 - mx);
                float e3 = (lane < 4) ? __expf(s3 - mx) : 0.f;
                float sum = e0 + e1 + e2 + e3;
#pragma unroll
                for (int off = 16; off >= 1; off >>= 1)
                    sum += __shfl_xor(sum, off, 32);
                float inv = 1.0f / sum;
                float* ab = &sAtt[wid * 104];
                ab[lane]      = e0 * inv;
                ab[lane + 32] = e1 * inv;
                ab[lane + 64] = e2 * inv;
                if (lane < 4) ab[lane + 96] = e3 * inv;
            }
        }
        __syncthreads();

        // ---- y[h][i] = sum_s att[h][s] * x[s][i]  -> LDS Y[sample] ----
#pragma unroll
        for (int pi = 0; pi < 2; ++pi) {
            int p = t + 256 * pi;          // 0..511
            int h = p >> 7;
            int i = p & 127;
            const float* ab = &sAtt[h * 104];
            float acc = 0.f;
            for (int s = 0; s < SPLIT; ++s)
                acc = fmaf(ab[s], Xb[s * XSTRIDE + i], acc);
            sY[k * YSTRIDE + p] = acc;
        }
    }
    __syncthreads();

    // ---- Phase 2: out[16 samples][h][64] = Y(16x128) @ W_h(128x64) ----
    // exact fp32 via V_WMMA_F32_16X16X4_F32; 16 tile jobs over 8 waves.
    {
        int wid  = t >> 5;
        int lane = t & 31;
        int m    = lane & 15;     // M (A) row / N (B,C) column within tile
        int half = lane >> 4;     // 0: lanes 0-15, 1: lanes 16-31
#pragma unroll
        for (int jj = 0; jj < 2; ++jj) {
            int j  = wid * 2 + jj;     // 0..15
            int h  = j >> 2;
            int jo = j & 3;
            int col = jo * 16 + m;
            const float* Wh = W + (size_t)h * IDIM * PERD;
            const float* Ym = &sY[m * YSTRIDE + h * IDIM];
            v8f acc = {0.f, 0.f, 0.f, 0.f, 0.f, 0.f, 0.f, 0.f};
#pragma unroll
            for (int k4 = 0; k4 < IDIM / 4; ++k4) {
                int kb = 4 * k4;
                v2f a, b;
                // A 16x4 f32: VGPR0 = K{0|2}, VGPR1 = K{1|3} (half selects)
                a.x = Ym[kb + 2 * half + 0];
                a.y = Ym[kb + 2 * half + 1];
                // B 4x16 f32: VGPR0 = row K{0|1}, VGPR1 = row K{2|3}
                b.x = Wh[(size_t)(kb + half) * PERD + col];
                b.y = Wh[(size_t)(kb + 2 + half) * PERD + col];
                acc = __builtin_amdgcn_wmma_f32_16x16x4_f32(
                        /*neg_a=*/false, a, /*neg_b=*/false, b,
                        /*c_mod=*/(short)0, acc,
                        /*reuse_a=*/false, /*reuse_b=*/false);
            }
            // C/D layout: VGPR r -> M = r (lanes 0-15) or r+8 (lanes 16-31)
#pragma unroll
            for (int r = 0; r < 8; ++r) {
                int samp = r + 8 * half;
                out[(size_t)(n0 + samp) * ODIM + h * PERD + col] = acc[r];
            }
        }
    }
}

// ---------------------------------------------------------------------------
// Host launcher
// ---------------------------------------------------------------------------
extern "C" void kernel_launch(void* const* d_in, const int* in_sizes, int n_in,
                              void* d_out, int out_size, void* d_ws, size_t ws_size,
                              hipStream_t stream) {
    const float* x = (const float*)d_in[0];   // [16384, 12800] f32
    const float* W = (const float*)d_in[1];   // [4, 128, 64]   f32
    const float* q = (const float*)d_in[2];   // [4, 64]        f32
    float* out = (float*)d_out;               // [16384, 256]   f32
    float* v   = (float*)d_ws;                // 4*128 floats scratch

    compute_v_kernel<<<1, 512, 0, stream>>>(W, q, v);

    int nwg = NTOT / SAMP_PER_WG;             // 1024
    attn_pool_kernel<<<nwg, THREADS, SM_BYTES, stream>>>(x, W, v, out);
}